// TriInter_317827579942
// MI455X (gfx1250) — compile-verified
//
#include <hip/hip_runtime.h>

#define HIDDEN   512
#define H1       513            // HIDDEN + 1
#define OUTDIM   256
#define BATCH    128
#define KPAD     516            // H1 rounded up to multiple of 4 (WMMA K-step)
#define VH_STRIDE 520
#define AH_STRIDE 528           // 33 * 16
#define NTILES_I 33             // ceil(513 / 16)
#define EPS      1e-5f

typedef __attribute__((ext_vector_type(2))) float v2f;
typedef __attribute__((ext_vector_type(4))) float v4f;
typedef __attribute__((ext_vector_type(8))) float v8f;

// ---------------------------------------------------------------------------
// Kernel 1: per-batch stats (mu, 1/sigma), build padded ah/vh rows, and zero
// the S1/S2/S3 accumulators (must re-zero every call: harness replays graph).
// ---------------------------------------------------------------------------
__global__ __launch_bounds__(256) void prep_kernel(
    const float* __restrict__ a, const float* __restrict__ v,
    float* __restrict__ vhbuf, float* __restrict__ ahbuf,
    float* __restrict__ mu, float* __restrict__ rsig,
    float* __restrict__ S1, float* __restrict__ S2, float* __restrict__ S3)
{
    const int b = blockIdx.x;
    const int t = threadIdx.x;
    const float* ar = a + b * HIDDEN;
    const float* vr = v + b * HIDDEN;

    float sa = 0.f, sv = 0.f, sa2 = 0.f, sv2 = 0.f;
    for (int i = t; i < HIDDEN; i += 256) {
        float x = ar[i], y = vr[i];
        sa += x; sa2 = fmaf(x, x, sa2);
        sv += y; sv2 = fmaf(y, y, sv2);
        ahbuf[b * AH_STRIDE + 1 + i] = x;
        vhbuf[b * VH_STRIDE + 1 + i] = y;
    }
    if (t == 0) { ahbuf[b * AH_STRIDE] = 1.f; vhbuf[b * VH_STRIDE] = 1.f; }
    for (int i = H1 + t; i < AH_STRIDE; i += 256) ahbuf[b * AH_STRIDE + i] = 0.f;
    for (int i = H1 + t; i < VH_STRIDE; i += 256) vhbuf[b * VH_STRIDE + i] = 0.f;

    // zero accumulators (each element written by exactly one thread)
    S1[b * OUTDIM + t] = 0.f;
    if (b == 0) { S2[t] = 0.f; S3[t] = 0.f; }

    // wave32 reduction, then cross-wave via LDS
    for (int off = 16; off > 0; off >>= 1) {
        sa  += __shfl_xor(sa,  off, 32);
        sv  += __shfl_xor(sv,  off, 32);
        sa2 += __shfl_xor(sa2, off, 32);
        sv2 += __shfl_xor(sv2, off, 32);
    }
    __shared__ float red[4][8];
    const int wave = t >> 5;
    if ((t & 31) == 0) { red[0][wave]=sa; red[1][wave]=sv; red[2][wave]=sa2; red[3][wave]=sv2; }
    __syncthreads();
    if (t == 0) {
        float A=0.f, V=0.f, A2=0.f, V2=0.f;
        for (int w = 0; w < 8; ++w) { A+=red[0][w]; V+=red[1][w]; A2+=red[2][w]; V2+=red[3][w]; }
        A += 1.f; V += 1.f; A2 += 1.f; V2 += 1.f;   // prepended ones
        const float invD = 1.f / (float)((size_t)H1 * H1);
        float m   = A * V * invD;
        float e2  = A2 * V2 * invD;
        float var = e2 - m * m;
        mu[b]   = m;
        rsig[b] = rsqrtf(var + EPS);
    }
}

// ---------------------------------------------------------------------------
// Kernel 2: one streaming pass over W. Block = (o, i-tile of 16). Loads
// 16x516 tile of (gamma .* W) into LDS (accumulating S2/S3 partials), then
// 8 waves each compute a 16(i) x 16(b) fp32 WMMA tile over K=513 against vh,
// contract with ah, and atomically accumulate S1[b,o].
// ---------------------------------------------------------------------------
__global__ __launch_bounds__(256) void fused_wmma_kernel(
    const float* __restrict__ W, const float* __restrict__ gamma,
    const float* __restrict__ beta, const float* __restrict__ vhbuf,
    const float* __restrict__ ahbuf, float* __restrict__ S1,
    float* __restrict__ S2, float* __restrict__ S3)
{
    __shared__ float ldsA[16 * KPAD];   // 33 KB: gamma-scaled W tile, zero-padded

    const int itile = blockIdx.x;       // 0..32
    const int o     = blockIdx.y;       // 0..255
    const int i0    = itile * 16;
    const int tid   = threadIdx.x;
    const size_t DD = (size_t)H1 * H1;
    const float* Wrow = W + (size_t)o * DD;

    // Cooperative load: W tile * gamma -> LDS; fold in S2/S3 partial sums.
    float s2p = 0.f, s3p = 0.f;
    for (int idx = tid; idx < 16 * KPAD; idx += 256) {
        const int il = idx / KPAD;
        const int j  = idx - il * KPAD;
        const int i  = i0 + il;
        float wg = 0.f;
        if (i < H1 && j < H1) {
            const size_t d = (size_t)i * H1 + j;
            const float wv = Wrow[d];
            wg  = wv * gamma[d];
            s2p += wg;
            s3p = fmaf(beta[d], wv, s3p);
        }
        ldsA[idx] = wg;
    }
    for (int off = 16; off > 0; off >>= 1) {
        s2p += __shfl_xor(s2p, off, 32);
        s3p += __shfl_xor(s3p, off, 32);
    }
    if ((tid & 31) == 0) { atomicAdd(&S2[o], s2p); atomicAdd(&S3[o], s3p); }
    __syncthreads();

    // WMMA phase: wave w handles batch columns [16w, 16w+16).
    const int wave = tid >> 5;
    const int lane = tid & 31;
    const int hi   = lane >> 4;     // 0: K+{0,1} / M rows 0..7 of acc; 1: K+{2,3} / rows 8..15
    const int lb   = lane & 15;
    const int b    = wave * 16 + lb;

    const float* __restrict__ vrow = vhbuf + (size_t)b * VH_STRIDE + 2 * hi;

    v8f acc = {};
    #pragma unroll 4
    for (int k0 = 0; k0 < KPAD; k0 += 4) {
        // A (16x4 f32): lane<16 -> row lb, K=k0..k0+1 ; lane>=16 -> row lb, K=k0+2..k0+3
        v2f av = *(const v2f*)&ldsA[lb * KPAD + 2 * hi + k0];
        // B (4x16 f32): lane<16 -> K=k0..k0+1, N=lb ; lane>=16 -> K=k0+2..k0+3, N=lb
        v2f bv = *(const v2f*)&vrow[k0];
        acc = __builtin_amdgcn_wmma_f32_16x16x4_f32(
            false, av, false, bv, (short)0, acc, false, false);
    }

    // acc VGPR g holds (i_local = g + 8*hi, b). Contract with ah over i_local.
    const float* ap = ahbuf + (size_t)b * AH_STRIDE + i0 + 8 * hi;
    v4f a0 = *(const v4f*)(ap);
    v4f a1 = *(const v4f*)(ap + 4);
    float p = 0.f;
    p = fmaf(a0.x, acc[0], p);
    p = fmaf(a0.y, acc[1], p);
    p = fmaf(a0.z, acc[2], p);
    p = fmaf(a0.w, acc[3], p);
    p = fmaf(a1.x, acc[4], p);
    p = fmaf(a1.y, acc[5], p);
    p = fmaf(a1.z, acc[6], p);
    p = fmaf(a1.w, acc[7], p);

    // lanes L and L^16 share the same b; sum covers i_local 0..15
    const float ptot = p + __shfl_xor(p, 16, 32);
    if (hi == 0) atomicAdd(&S1[b * OUTDIM + o], ptot);
}

// ---------------------------------------------------------------------------
// Kernel 3: out[b,o] = relu( rsig_b*(S1 - mu_b*S2) + S3 + bias )
// ---------------------------------------------------------------------------
__global__ __launch_bounds__(256) void finalize_kernel(
    const float* __restrict__ S1, const float* __restrict__ S2,
    const float* __restrict__ S3, const float* __restrict__ bias,
    const float* __restrict__ mu, const float* __restrict__ rsig,
    float* __restrict__ out)
{
    const int b = blockIdx.x;
    const int o = threadIdx.x;
    const int idx = b * OUTDIM + o;
    const float val = rsig[b] * (S1[idx] - mu[b] * S2[o]) + S3[o] + bias[o];
    out[idx] = fmaxf(val, 0.f);
}

// ---------------------------------------------------------------------------
extern "C" void kernel_launch(void* const* d_in, const int* in_sizes, int n_in,
                              void* d_out, int out_size, void* d_ws, size_t ws_size,
                              hipStream_t stream) {
    const float* a     = (const float*)d_in[0];   // [128, 512]
    const float* v     = (const float*)d_in[1];   // [128, 512]
    const float* gamma = (const float*)d_in[2];   // [D]
    const float* beta  = (const float*)d_in[3];   // [D]
    const float* W     = (const float*)d_in[4];   // [256, D]
    const float* bias  = (const float*)d_in[5];   // [256]
    float* out = (float*)d_out;                   // [128, 256]

    float* ws    = (float*)d_ws;
    float* vhbuf = ws;                               // 128 * 520
    float* ahbuf = vhbuf + BATCH * VH_STRIDE;        // 128 * 528
    float* mu    = ahbuf + BATCH * AH_STRIDE;        // 128
    float* rsig  = mu + BATCH;                       // 128
    float* S1    = rsig + BATCH;                     // 128 * 256
    float* S2    = S1 + BATCH * OUTDIM;              // 256
    float* S3    = S2 + OUTDIM;                      // 256

    prep_kernel<<<BATCH, 256, 0, stream>>>(a, v, vhbuf, ahbuf, mu, rsig, S1, S2, S3);

    dim3 grid(NTILES_I, OUTDIM);
    fused_wmma_kernel<<<grid, 256, 0, stream>>>(W, gamma, beta, vhbuf, ahbuf, S1, S2, S3);

    finalize_kernel<<<BATCH, 256, 0, stream>>>(S1, S2, S3, bias, mu, rsig, out);
}